// ChildSumTreeLSTM_7078106104355
// MI455X (gfx1250) — compile-verified
//
#include <hip/hip_runtime.h>
#include <hip/hip_bf16.h>

typedef __attribute__((ext_vector_type(2))) float v2f;
typedef __attribute__((ext_vector_type(8))) float v8f;
typedef __attribute__((ext_vector_type(4))) int   v4i;

#define NN      8192      // nodes
#define MD      1024      // mem dim
#define KD      1024      // inner-K of every GEMM in this model

#define BM      64
#define BN      128
#define BK      32
#define LSTR    36        // LDS row stride (floats): 144B = 16B-aligned, conflict-free

#if defined(__HIP_DEVICE_COMPILE__) && __has_builtin(__builtin_amdgcn_global_load_async_to_lds_b128)
#define ASYNC_STAGE 1
#else
#define ASYNC_STAGE 0
#endif

template <int N>
__device__ __forceinline__ void wait_async() {
#if __has_builtin(__builtin_amdgcn_s_wait_asynccnt)
    __builtin_amdgcn_s_wait_asynccnt(N);
#else
    asm volatile("s_wait_asynccnt %0" ::"i"(N) : "memory");
#endif
}

#if ASYNC_STAGE
// 16B global -> LDS DMA, tracked by ASYNCcnt. Integer round-trips give us the
// correctly-typed addrspace pointers: flat global == AS(1) numerically; flat
// LDS low 32 bits == AS(3) offset (ISA 10.2 aperture rules; p3 is 32-bit).
__device__ __forceinline__ void async_b128(const float* g, const float* l) {
    __builtin_amdgcn_global_load_async_to_lds_b128(
        (__attribute__((address_space(1))) v4i*)(unsigned long long)g,
        (__attribute__((address_space(3))) v4i*)(unsigned int)(unsigned long long)l,
        0, 0);
}
#endif

__device__ __forceinline__ float sigmoidf_(float x) {
    return 1.0f / (1.0f + __expf(-x));
}

// ---------------------------------------------------------------------------
// Generic K=1024 WMMA GEMM (f32, V_WMMA_F32_16X16X4_F32):
//   OUT[r][outColOff + c] = sum_k A[row(r)][k] * W[c][k]
//                           (+ bias[c]) (+ extra[extraLo + (r>>extraShift)][c])
//                           (sigmoid if requested)
// row(r) = rowIdx ? rowIdx[r] : r  (gather rows, e.g. child-c rows).
// Block: 256 threads (8 waves), 64x128 tile, BK=32, double-buffered LDS
// filled with GLOBAL_LOAD_ASYNC_TO_LDS_B128 (ASYNCcnt) when available.
// Each wave owns a 32x32 subtile = 4 accumulators of 16x16.
// NOTE: A-rows beyond M_ are loaded unguarded (provably in-bounds for every
// call site) and only feed output rows that the store guard discards.
// ---------------------------------------------------------------------------
__global__ __launch_bounds__(256)
void wmma_gemm_k1024(const float* __restrict__ A,
                     const int*   __restrict__ rowIdx,
                     int M_,
                     const float* __restrict__ W,        // [NCOLS][1024] row-major
                     const float* __restrict__ bias,     // [NCOLS] or null
                     const float* __restrict__ extra,    // per-row addend or null
                     int extraLo, int extraShift, int extraStride,
                     float* __restrict__ OUT, int outStride, int outColOff,
                     int applySigmoid)
{
    __shared__ __align__(16) float As[2][BM * LSTR];
    __shared__ __align__(16) float Ws[2][BN * LSTR];

    const int tid  = threadIdx.x;
    const int lane = tid & 31;
    const int wid  = tid >> 5;
    const int l    = lane & 15;    // M (A) / N (B) index within a 16-tile
    const int half = lane >> 4;    // K-half selector per ISA f32 frag layout
    const int mT   = (wid & 1) * 32;
    const int nT   = (wid >> 1) * 32;
    const int rowBase = blockIdx.x * BM;
    const int colBase = blockIdx.y * BN;

    // ---- per-thread staging coordinates (resolved once) ----
    long aOff[2];   // element offset of this thread's A float4 (col 0 of chunk)
    int  aLds[2];
    #pragma unroll
    for (int it = 0; it < 2; ++it) {              // A: 64x32 -> 512 float4 slots
        int slot = tid + it * 256;
        int r    = slot >> 3;
        int cg   = (slot & 7) * 4;
        int gr   = rowBase + r;
        long ar  = rowIdx ? (long)rowIdx[gr] : (long)gr;
        aOff[it] = ar * (long)KD + cg;
        aLds[it] = r * LSTR + cg;
    }
    long wOff[4];
    int  wLds[4];
    #pragma unroll
    for (int it = 0; it < 4; ++it) {              // W: 128x32 -> 1024 float4 slots
        int slot = tid + it * 256;
        int r    = slot >> 3;
        int cg   = (slot & 7) * 4;
        wOff[it] = (long)(colBase + r) * KD + cg;
        wLds[it] = r * LSTR + cg;
    }

    auto stage = [&](int buf, int kk) {
#if ASYNC_STAGE
        #pragma unroll
        for (int it = 0; it < 2; ++it)
            async_b128(A + aOff[it] + kk, &As[buf][aLds[it]]);
        #pragma unroll
        for (int it = 0; it < 4; ++it)
            async_b128(W + wOff[it] + kk, &Ws[buf][wLds[it]]);
#else
        #pragma unroll
        for (int it = 0; it < 2; ++it) {
            float4 v = *(const float4*)(A + aOff[it] + kk);
            float* p = &As[buf][aLds[it]];
            p[0] = v.x; p[1] = v.y; p[2] = v.z; p[3] = v.w;
        }
        #pragma unroll
        for (int it = 0; it < 4; ++it) {
            float4 v = *(const float4*)(W + wOff[it] + kk);
            float* p = &Ws[buf][wLds[it]];
            p[0] = v.x; p[1] = v.y; p[2] = v.z; p[3] = v.w;
        }
#endif
    };

    v8f acc00 = {}, acc01 = {}, acc10 = {}, acc11 = {};

    stage(0, 0);
    for (int kk = 0; kk < KD; kk += BK) {
        int cur = (kk >> 5) & 1;
        bool more = (kk + BK) < KD;
        if (more) stage(cur ^ 1, kk + BK);        // prefetch into other buffer
#if ASYNC_STAGE
        if (more) wait_async<6>();                // cur's 6 done (in-order), next in flight
        else      wait_async<0>();
#endif
        __syncthreads();

        #pragma unroll
        for (int ks = 0; ks < BK; ks += 4) {
            int kc = ks + 2 * half;               // lanes 0-15: K%4 {0,1}; 16-31: {2,3}
            const float* ap0 = &As[cur][(mT + l) * LSTR + kc];
            const float* ap1 = &As[cur][(mT + 16 + l) * LSTR + kc];
            const float* bp0 = &Ws[cur][(nT + l) * LSTR + kc];
            const float* bp1 = &Ws[cur][(nT + 16 + l) * LSTR + kc];
            v2f a0, a1, b0, b1;
            a0.x = ap0[0]; a0.y = ap0[1];
            a1.x = ap1[0]; a1.y = ap1[1];
            b0.x = bp0[0]; b0.y = bp0[1];
            b1.x = bp1[0]; b1.y = bp1[1];
            acc00 = __builtin_amdgcn_wmma_f32_16x16x4_f32(false, a0, false, b0, (short)0, acc00, false, false);
            acc01 = __builtin_amdgcn_wmma_f32_16x16x4_f32(false, a0, false, b1, (short)0, acc01, false, false);
            acc10 = __builtin_amdgcn_wmma_f32_16x16x4_f32(false, a1, false, b0, (short)0, acc10, false, false);
            acc11 = __builtin_amdgcn_wmma_f32_16x16x4_f32(false, a1, false, b1, (short)0, acc11, false, false);
        }
        __syncthreads();
    }

    // ---- fused epilogue + store (C/D layout: vgpr i -> M = i + 8*half, N = l) ----
    auto storeTile = [&](v8f acc, int mi, int ni) {
        int c = colBase + nT + ni * 16 + l;
        #pragma unroll
        for (int i = 0; i < 8; ++i) {
            int r = rowBase + mT + mi * 16 + i + 8 * half;
            if (r < M_) {
                float v = acc[i];
                if (bias)  v += bias[c];
                if (extra) v += extra[(long)(extraLo + (r >> extraShift)) * (long)extraStride + c];
                if (applySigmoid) v = sigmoidf_(v);
                OUT[(long)r * outStride + outColOff + c] = v;
            }
        }
    };
    storeTile(acc00, 0, 0);
    storeTile(acc01, 0, 1);
    storeTile(acc10, 1, 0);
    storeTile(acc11, 1, 1);
}

// ---------------------------------------------------------------------------
// Zero the pad row (index NN) of h_buf / c_buf. 1024 threads total.
// ---------------------------------------------------------------------------
__global__ void zero_pad_rows(float* __restrict__ h_buf, float* __restrict__ c_buf) {
    int i = blockIdx.x * blockDim.x + threadIdx.x;
    if (i < MD) {
        h_buf[(size_t)NN * MD + i] = 0.f;
        c_buf[(size_t)NN * MD + i] = 0.f;
    }
}

// ---------------------------------------------------------------------------
// Leaves (t = 1024..8191): children are all pad -> csum=0, fc=0.
//   iou = XW[t][0:3072] + b_iouh ; c = sig(i)*tanh(u) ; h = sig(o)*tanh(c)
// ---------------------------------------------------------------------------
__global__ __launch_bounds__(256)
void leaf_kernel(const float* __restrict__ XW, const float* __restrict__ b_iouh,
                 float* __restrict__ h_buf, float* __restrict__ c_buf)
{
    int t = 1024 + blockIdx.x;
    const float* row = XW + (size_t)t * 4096;
    #pragma unroll
    for (int k = 0; k < 4; ++k) {
        int m = threadIdx.x + k * 256;
        float ig = sigmoidf_(row[m]        + b_iouh[m]);
        float og = sigmoidf_(row[1024 + m] + b_iouh[1024 + m]);
        float ug = tanhf    (row[2048 + m] + b_iouh[2048 + m]);
        float c  = ig * ug;
        c_buf[(size_t)t * MD + m] = c;
        h_buf[(size_t)t * MD + m] = og * tanhf(c);
    }
}

// ---------------------------------------------------------------------------
// csum[i] = sum_j c_buf[children[lo+i][j]]   (pad index NN hits the zero row)
// ---------------------------------------------------------------------------
__global__ __launch_bounds__(256)
void csum_kernel(const float* __restrict__ c_buf, const int* __restrict__ children,
                 int lo, float* __restrict__ csum)
{
    int i = blockIdx.x;
    const int* ch = children + (size_t)(lo + i) * 8;
    #pragma unroll
    for (int k = 0; k < 4; ++k) {
        int m = threadIdx.x + k * 256;
        float s = 0.f;
        #pragma unroll
        for (int j = 0; j < 8; ++j)
            s += c_buf[(size_t)ch[j] * MD + m];
        csum[(size_t)i * MD + m] = s;
    }
}

// ---------------------------------------------------------------------------
// Combine: gates from tiou (already = iou_x + csum@w_iouh^T + b_iouh),
// f already sigmoided in tf; fc multiplies the children's *h* rows
// (faithful to the reference's swapped usage).
// ---------------------------------------------------------------------------
__global__ __launch_bounds__(256)
void combine_kernel(const float* __restrict__ tiou, const float* __restrict__ tf,
                    const int* __restrict__ children, int lo,
                    float* __restrict__ h_buf, float* __restrict__ c_buf)
{
    int i = blockIdx.x;
    int t = lo + i;
    const int* ch = children + (size_t)t * 8;
    #pragma unroll
    for (int k = 0; k < 4; ++k) {
        int m = threadIdx.x + k * 256;
        float ig = sigmoidf_(tiou[(size_t)i * 3072 + m]);
        float og = sigmoidf_(tiou[(size_t)i * 3072 + 1024 + m]);
        float ug = tanhf    (tiou[(size_t)i * 3072 + 2048 + m]);
        float fc = 0.f;
        #pragma unroll
        for (int j = 0; j < 8; ++j)
            fc += tf[(size_t)(i * 8 + j) * MD + m] * h_buf[(size_t)ch[j] * MD + m];
        float c = ig * ug + fc;
        c_buf[(size_t)t * MD + m] = c;
        h_buf[(size_t)t * MD + m] = og * tanhf(c);
    }
}

// ---------------------------------------------------------------------------
// out = [root_h (1024) | root_c (1024)]
// ---------------------------------------------------------------------------
__global__ void copyout_kernel(const float* __restrict__ h_buf,
                               const float* __restrict__ c_buf,
                               float* __restrict__ out)
{
    int m = blockIdx.x * 256 + threadIdx.x;   // 0..2047
    if (m < MD) out[m] = h_buf[m];
    else        out[m] = c_buf[m - MD];
}

// ---------------------------------------------------------------------------
extern "C" void kernel_launch(void* const* d_in, const int* in_sizes, int n_in,
                              void* d_out, int out_size, void* d_ws, size_t ws_size,
                              hipStream_t stream)
{
    (void)in_sizes; (void)n_in; (void)out_size; (void)ws_size;

    const float* inputs   = (const float*)d_in[0];
    const int*   children = (const int*)  d_in[1];
    const float* w_ioux   = (const float*)d_in[2];
    const float* b_ioux   = (const float*)d_in[3];
    const float* w_iouh   = (const float*)d_in[4];
    const float* b_iouh   = (const float*)d_in[5];
    const float* w_fx     = (const float*)d_in[6];
    const float* b_fx     = (const float*)d_in[7];
    const float* w_fh     = (const float*)d_in[8];
    const float* b_fh     = (const float*)d_in[9];
    float* out = (float*)d_out;
    float* ws  = (float*)d_ws;

    // Workspace layout (floats)
    float* XW    = ws;                               // [8192][4096]  iou_x | fx_x
    float* h_buf = XW    + (size_t)NN * 4096;        // [8193][1024]  (+pad row)
    float* c_buf = h_buf + (size_t)(NN + 1) * MD;    // [8193][1024]
    float* csum  = c_buf + (size_t)(NN + 1) * MD;    // [512][1024]
    float* tiou  = csum  + (size_t)512 * MD;         // [512][3072]
    float* tf    = tiou  + (size_t)512 * 3072;       // [4096][1024]

    zero_pad_rows<<<4, 256, 0, stream>>>(h_buf, c_buf);

    // Phase 1: big input-side GEMMs (fused bias), one per weight matrix.
    wmma_gemm_k1024<<<dim3(NN / BM, 3072 / BN), 256, 0, stream>>>(
        inputs, nullptr, NN, w_ioux, b_ioux,
        nullptr, 0, 0, 0, XW, 4096, 0, 0);
    wmma_gemm_k1024<<<dim3(NN / BM, 1024 / BN), 256, 0, stream>>>(
        inputs, nullptr, NN, w_fx, b_fx,
        nullptr, 0, 0, 0, XW, 4096, 3072, 0);

    // Phase 2: all 7168 leaves in one parallel pass (no matmul needed).
    leaf_kernel<<<NN - 1024, 256, 0, stream>>>(XW, b_iouh, h_buf, c_buf);

    // Phase 3: internal levels, leaves-first (children have larger indices).
    const int levLo[5] = {585, 73, 9, 1, 0};
    const int levN [5] = {439, 512, 64, 8, 1};
    for (int s = 0; s < 5; ++s) {
        int lo = levLo[s], n = levN[s];
        // csum over children's c rows
        csum_kernel<<<n, 256, 0, stream>>>(c_buf, children, lo, csum);
        // tiou[i] = iou_x[lo+i] + csum[i] @ w_iouh^T + b_iouh
        wmma_gemm_k1024<<<dim3((n + BM - 1) / BM, 3072 / BN), 256, 0, stream>>>(
            csum, nullptr, n, w_iouh, b_iouh,
            XW, lo, 0, 4096, tiou, 3072, 0, 0);
        // tf[i*8+j] = sigmoid(c_buf[child(i,j)] @ w_fh^T + b_fh + fx_x[lo+i])
        wmma_gemm_k1024<<<dim3((n * 8 + BM - 1) / BM, 1024 / BN), 256, 0, stream>>>(
            c_buf, children + (size_t)lo * 8, n * 8, w_fh, b_fh,
            XW + 3072, lo, 3, 4096, tf, 1024, 0, 1);
        // gates + fc combine -> h_buf[t], c_buf[t]
        combine_kernel<<<n, 256, 0, stream>>>(tiou, tf, children, lo, h_buf, c_buf);
    }

    copyout_kernel<<<8, 256, 0, stream>>>(h_buf, c_buf, out);
}